// GAT_41248865911241
// MI455X (gfx1250) — compile-verified
//
#include <hip/hip_runtime.h>
#include <math.h>

// Problem constants (match reference)
#define FDIM   128
#define HEADS  4
#define DDIM   128
#define GGR    64
#define NEG_SLOPE 0.2f
#define BN_EPS    1e-5f

typedef __attribute__((ext_vector_type(2))) float v2f;
typedef __attribute__((ext_vector_type(8))) float v8f;

// ---------------------------------------------------------------- utilities
__global__ void fill_kernel(float* __restrict__ p, float v, long long n) {
  long long i = (long long)blockIdx.x * blockDim.x + threadIdx.x;
  if (i < n) p[i] = v;
}

// ------------------------------------------------ WMMA GEMM  C = A[N,128] * B[128,128]
// One block = 16 output rows; 8 waves each own one 16-wide N tile.
// Uses V_WMMA_F32_16X16X4_F32 (fp32-exact, wave32 layout per CDNA5 ISA 7.12.2).
__global__ __launch_bounds__(256) void gemm128_wmma(const float* __restrict__ A,
                                                    const float* __restrict__ B,
                                                    float* __restrict__ C) {
  __shared__ float As[16 * 128];  // 8 KB A tile, reused by all 8 waves
  const int m_base = blockIdx.x * 16;
  const int tid = threadIdx.x;
  for (int i = tid; i < 16 * 128; i += 256) {
    int r = i >> 7, c = i & 127;
    As[i] = A[(long long)(m_base + r) * FDIM + c];
  }
  __syncthreads();

  const int wave = tid >> 5;
  const int lane = tid & 31;
  const int n_base = wave * 16;
  const int half = lane >> 4;      // 0: lanes 0-15 (K 0,1) | 1: lanes 16-31 (K 2,3)
  const int lm = lane & 15;        // A: row M ; B: col N
  const int koff = half * 2;

  v8f acc = {};
  for (int kk = 0; kk < 128; kk += 4) {
    v2f a, b;
    a.x = As[lm * 128 + kk + koff];
    a.y = As[lm * 128 + kk + koff + 1];
    b.x = B[(long long)(kk + koff) * 128 + n_base + lm];
    b.y = B[(long long)(kk + koff + 1) * 128 + n_base + lm];
    acc = __builtin_amdgcn_wmma_f32_16x16x4_f32(false, a, false, b,
                                                (short)0, acc, false, false);
  }
  // D layout: VGPR r -> M=r (lanes 0-15) / M=r+8 (lanes 16-31), N = lane&15
  for (int r = 0; r < 8; ++r) {
    int M = r + half * 8;
    C[(long long)(m_base + M) * DDIM + n_base + lm] = acc[r];
  }
}

// ------------------------------------------------ attention scores  a_s,a_d [N,H]
__global__ __launch_bounds__(128) void att_scores(const float* __restrict__ h,
                                                  const float* __restrict__ att_s,
                                                  const float* __restrict__ att_d,
                                                  float* __restrict__ as,
                                                  float* __restrict__ ad) {
  const int node = blockIdx.x;
  const int t = threadIdx.x;               // dim 0..127; head = t>>5 aligns with wave
  float v = h[(long long)node * DDIM + t];
  float s = v * att_s[t];
  float d = v * att_d[t];
  for (int o = 16; o > 0; o >>= 1) {
    s += __shfl_xor(s, o, 32);
    d += __shfl_xor(d, o, 32);
  }
  if ((t & 31) == 0) {
    as[(long long)node * HEADS + (t >> 5)] = s;
    ad[(long long)node * HEADS + (t >> 5)] = d;
  }
}

__device__ __forceinline__ void edge_pair(const int* __restrict__ ei, long long E,
                                          long long e, int& s, int& d) {
  if (e < E) { s = ei[e]; d = ei[E + e]; }
  else       { s = d = (int)(e - E); }     // synthesized self-loop
}

__device__ __forceinline__ float lrelu_logit(const float* as, const float* ad,
                                             int s, int d, int hh) {
  float v = as[(long long)s * HEADS + hh] + ad[(long long)d * HEADS + hh];
  return v > 0.f ? v : NEG_SLOPE * v;
}

// ------------------------------------------------ segment max over dst
__global__ void edge_max(const int* __restrict__ ei, long long E, long long etot,
                         const float* __restrict__ as, const float* __restrict__ ad,
                         float* __restrict__ m) {
  long long gid = (long long)blockIdx.x * blockDim.x + threadIdx.x;
  if (gid >= etot * HEADS) return;
  long long e = gid >> 2; int hh = (int)(gid & 3);
  int s, d; edge_pair(ei, E, e, s, d);
  atomicMax(&m[(long long)d * HEADS + hh], lrelu_logit(as, ad, s, d, hh));
}

// ------------------------------------------------ segment sum of exp
__global__ void edge_denom(const int* __restrict__ ei, long long E, long long etot,
                           const float* __restrict__ as, const float* __restrict__ ad,
                           const float* __restrict__ m, float* __restrict__ denom) {
  long long gid = (long long)blockIdx.x * blockDim.x + threadIdx.x;
  if (gid >= etot * HEADS) return;
  long long e = gid >> 2; int hh = (int)(gid & 3);
  int s, d; edge_pair(ei, E, e, s, d);
  float v = lrelu_logit(as, ad, s, d, hh);
  atomicAdd(&denom[(long long)d * HEADS + hh], expf(v - m[(long long)d * HEADS + hh]));
}

// ------------------------------------------------ weighted scatter-add (dominant pass)
// thread = (edge, dim): warps issue 128B contiguous loads of h[src] and atomics on agg[dst]
__global__ void edge_agg(const int* __restrict__ ei, long long E, long long etot,
                         const float* __restrict__ as, const float* __restrict__ ad,
                         const float* __restrict__ m, const float* __restrict__ denom,
                         const float* __restrict__ h, float* __restrict__ agg) {
  long long gid = (long long)blockIdx.x * blockDim.x + threadIdx.x;
  if (gid >= etot * DDIM) return;
  long long e = gid >> 7; int dd = (int)(gid & 127); int hh = dd >> 5;
  int s, d; edge_pair(ei, E, e, s, d);
  float v = lrelu_logit(as, ad, s, d, hh);
  long long mi = (long long)d * HEADS + hh;
  float alpha = expf(v - m[mi]) / (denom[mi] + 1e-16f);
  atomicAdd(&agg[(long long)d * DDIM + dd],
            h[(long long)s * DDIM + dd] * alpha);
}

// ------------------------------------------------ bias + BN(eval) + ReLU
__global__ void post_bn_relu(const float* __restrict__ agg, const float* __restrict__ bias,
                             const float* __restrict__ g, const float* __restrict__ b,
                             const float* __restrict__ mean, const float* __restrict__ var,
                             float* __restrict__ out, long long total) {
  long long i = (long long)blockIdx.x * blockDim.x + threadIdx.x;
  if (i >= total) return;
  int dd = (int)(i & 127);
  float val = agg[i] + bias[dd];
  val = (val - mean[dd]) * rsqrtf(var[dd] + BN_EPS) * g[dd] + b[dd];
  out[i] = val > 0.f ? val : 0.f;
}

// ------------------------------------------------ mean pool (sum + counts)
__global__ void pool_kernel(const float* __restrict__ act, const int* __restrict__ batch,
                            float* __restrict__ pooled, float* __restrict__ counts,
                            long long total) {
  long long i = (long long)blockIdx.x * blockDim.x + threadIdx.x;
  if (i >= total) return;
  long long node = i >> 7; int dd = (int)(i & 127);
  int gidx = batch[node];
  atomicAdd(&pooled[(long long)gidx * DDIM + dd], act[i]);
  if (dd == 0) atomicAdd(&counts[gidx], 1.0f);
}

// ------------------------------------------------ tiny MLP head: [64,128]->[64,1]
__global__ __launch_bounds__(64) void head_mlp(const float* __restrict__ pooled,
                                               const float* __restrict__ counts,
                                               const float* __restrict__ w1,
                                               const float* __restrict__ b1,
                                               const float* __restrict__ w2,
                                               const float* __restrict__ b2,
                                               float* __restrict__ out) {
  __shared__ float z[64];
  const int gg = blockIdx.x, j = threadIdx.x;
  float cnt = fmaxf(counts[gg], 1.0f);
  float acc = 0.f;
  for (int k = 0; k < 128; ++k)
    acc += (pooled[(long long)gg * DDIM + k] / cnt) * w1[k * 64 + j];
  acc += b1[j];
  z[j] = acc > 0.f ? acc : 0.f;
  __syncthreads();
  if (j == 0) {
    float o = b2[0];
    for (int k = 0; k < 64; ++k) o += z[k] * w2[k];
    out[gg] = o;
  }
}

// =====================================================================================
extern "C" void kernel_launch(void* const* d_in, const int* in_sizes, int n_in,
                              void* d_out, int out_size, void* d_ws, size_t ws_size,
                              hipStream_t stream) {
  const float* x        = (const float*)d_in[0];
  const int*   ei       = (const int*)d_in[1];
  const int*   batch    = (const int*)d_in[2];
  const float* w0       = (const float*)d_in[3];
  const float* att_s0   = (const float*)d_in[4];
  const float* att_d0   = (const float*)d_in[5];
  const float* bias0    = (const float*)d_in[6];
  const float* w1       = (const float*)d_in[7];
  const float* att_s1   = (const float*)d_in[8];
  const float* att_d1   = (const float*)d_in[9];
  const float* bias1    = (const float*)d_in[10];
  const float* bn_g0 = (const float*)d_in[11], *bn_b0 = (const float*)d_in[12];
  const float* bn_m0 = (const float*)d_in[13], *bn_v0 = (const float*)d_in[14];
  const float* bn_g1 = (const float*)d_in[15], *bn_b1 = (const float*)d_in[16];
  const float* bn_m1 = (const float*)d_in[17], *bn_v1 = (const float*)d_in[18];
  const float* lin1_w = (const float*)d_in[19], *lin1_b = (const float*)d_in[20];
  const float* lin2_w = (const float*)d_in[21], *lin2_b = (const float*)d_in[22];
  float* out = (float*)d_out;

  const long long N = in_sizes[0] / FDIM;
  const long long E = in_sizes[1] / 2;
  const long long ETOT = E + N;

  // workspace layout (floats)
  float* ws = (float*)d_ws;
  float* hW     = ws;                       // N*128 : gemm output
  float* agg    = hW     + N * DDIM;        // N*128 : aggregation
  float* act    = agg    + N * DDIM;        // N*128 : post-activation / next-layer input
  float* as_    = act    + N * DDIM;        // N*4
  float* ad_    = as_    + N * HEADS;       // N*4
  float* mseg   = ad_    + N * HEADS;       // N*4
  float* denom  = mseg   + N * HEADS;       // N*4
  float* pooled = denom  + N * HEADS;       // 64*128
  float* counts = pooled + GGR * DDIM;      // 64

  auto blocks = [](long long n, int bs) { return (unsigned)((n + bs - 1) / bs); };

  const long long nNH = N * HEADS;
  const long long nND = N * DDIM;
  const long long nEH = ETOT * HEADS;
  const long long nED = ETOT * DDIM;

  auto run_layer = [&](const float* in, const float* w,
                       const float* ats, const float* atd, const float* bias,
                       const float* g, const float* b, const float* mn, const float* vr) {
    gemm128_wmma<<<blocks(N, 16), 256, 0, stream>>>(in, w, hW);
    att_scores<<<(unsigned)N, 128, 0, stream>>>(hW, ats, atd, as_, ad_);
    fill_kernel<<<blocks(nNH, 256), 256, 0, stream>>>(mseg, -1e30f, nNH);
    fill_kernel<<<blocks(nNH, 256), 256, 0, stream>>>(denom, 0.f, nNH);
    edge_max<<<blocks(nEH, 256), 256, 0, stream>>>(ei, E, ETOT, as_, ad_, mseg);
    edge_denom<<<blocks(nEH, 256), 256, 0, stream>>>(ei, E, ETOT, as_, ad_, mseg, denom);
    fill_kernel<<<blocks(nND, 256), 256, 0, stream>>>(agg, 0.f, nND);
    edge_agg<<<blocks(nED, 256), 256, 0, stream>>>(ei, E, ETOT, as_, ad_, mseg, denom, hW, agg);
    post_bn_relu<<<blocks(nND, 256), 256, 0, stream>>>(agg, bias, g, b, mn, vr, act, nND);
  };

  run_layer(x,   w0, att_s0, att_d0, bias0, bn_g0, bn_b0, bn_m0, bn_v0);
  run_layer(act, w1, att_s1, att_d1, bias1, bn_g1, bn_b1, bn_m1, bn_v1);

  fill_kernel<<<blocks(GGR * DDIM + GGR, 256), 256, 0, stream>>>(pooled, 0.f, GGR * DDIM + GGR);
  pool_kernel<<<blocks(nND, 256), 256, 0, stream>>>(act, batch, pooled, counts, nND);
  head_mlp<<<GGR, 64, 0, stream>>>(pooled, counts, lin1_w, lin1_b, lin2_w, lin2_b, out);
}